// SMCLMDA_64063732187755
// MI455X (gfx1250) — compile-verified
//
#include <hip/hip_runtime.h>
#include <math.h>

typedef __attribute__((ext_vector_type(2))) float v2f;
typedef __attribute__((ext_vector_type(8))) float v8f;

#define TPB 256

// ---------------- utility kernels ----------------

__global__ void fill1_k(float* __restrict__ p, int n) {
  int i = blockIdx.x * blockDim.x + threadIdx.x;
  if (i < n) p[i] = 1.0f;
}

__global__ void zero_k(float* __restrict__ p, int n) {
  int i = blockIdx.x * blockDim.x + threadIdx.x;
  if (i < n) p[i] = 0.0f;
}

__global__ void deg_acc_k(const int* __restrict__ dst, const float* __restrict__ ew,
                          float* __restrict__ deg, int E) {
  int e = blockIdx.x * blockDim.x + threadIdx.x;
  if (e < E) unsafeAtomicAdd(&deg[dst[e]], ew[e]);
}

__global__ void rsqrt_k(float* __restrict__ p, int n) {
  int i = blockIdx.x * blockDim.x + threadIdx.x;
  if (i < n) { float d = p[i]; p[i] = (d > 0.f) ? rsqrtf(d) : 0.f; }
}

// ---------------- WMMA GEMM: H = X @ W^T  (X: nrows x 64, W: 64 x 64) ------
// fp32 exact math via V_WMMA_F32_16X16X4_F32; one 16x16 output tile per wave,
// 8 waves/block => 32 rows x 64 cols per block. K=64 -> 16 WMMA ops per tile.
__global__ void gemm_wmma_k(const float* __restrict__ X, const float* __restrict__ W,
                            float* __restrict__ H, int nrows) {
  const int lane    = threadIdx.x & 31;
  const int wave    = threadIdx.x >> 5;
  const int rowTile = blockIdx.x * 2 + (wave >> 2);
  const int colTile = wave & 3;
  const int m       = lane & 15;
  const int half    = lane >> 4;            // 0: K={0,1}, 1: K={2,3} per step

  int gr = rowTile * 16 + m;                // A row this lane supplies
  if (gr > nrows - 1) gr = nrows - 1;       // branchless clamp (EXEC stays full)
  const int gc = colTile * 16 + m;          // B column (== W row) this lane supplies

  const float* xrow = X + (size_t)gr * 64 + 2 * half;
  const float* wrow = W + (size_t)gc * 64 + 2 * half;   // B[k,o] = W[o,k]

  v8f acc = {};
#pragma unroll
  for (int k0 = 0; k0 < 64; k0 += 4) {
    v2f a = *(const v2f*)(xrow + k0);       // A[m, k0+2h], A[m, k0+2h+1]
    v2f b = *(const v2f*)(wrow + k0);       // B[k0+2h, n], B[k0+2h+1, n]
    acc = __builtin_amdgcn_wmma_f32_16x16x4_f32(
        /*neg_a=*/false, a, /*neg_b=*/false, b,
        /*c_mod=*/(short)0, acc, /*reuse_a=*/false, /*reuse_b=*/false);
  }

  // D layout: VGPR j of lane l holds (M = j + 8*(l>>4), N = l&15)
  const int n     = lane & 15;
  const int rbase = rowTile * 16 + 8 * half;
#pragma unroll
  for (int j = 0; j < 8; ++j) {
    int r = rbase + j;
    if (r < nrows) H[(size_t)r * 64 + colTile * 16 + n] = acc[j];
  }
}

// ---------------- GCN aggregation ----------------

// self-loop contribution: agg[n,:] = dinv[n]^2 * Hin[n,:]
__global__ void self_init_k(float* __restrict__ agg, const float* __restrict__ Hin,
                            const float* __restrict__ dinv, int n64) {
  int i = blockIdx.x * blockDim.x + threadIdx.x;
  if (i < n64) {
    float dv = dinv[i >> 6];
    agg[i] = dv * dv * Hin[i];
  }
}

// one wave32 per edge: gather 64 floats (float2/lane), scatter-add with
// hardware global_atomic_add_f32 (unsafeAtomicAdd avoids any CAS-loop fallback)
__global__ void edge_agg_k(const int* __restrict__ src, const int* __restrict__ dst,
                           const float* __restrict__ ew, const float* __restrict__ dinv,
                           const float* __restrict__ Hin, float* __restrict__ agg, int E) {
  const int lane = threadIdx.x & 31;
  const int e    = blockIdx.x * (blockDim.x >> 5) + (threadIdx.x >> 5);
  if (e >= E) return;
  const int   s    = src[e];
  const int   d    = dst[e];
  const float norm = dinv[s] * ew[e] * dinv[d];
  const float2 v   = ((const float2*)(Hin + (size_t)s * 64))[lane];
  float* ap = agg + (size_t)d * 64 + lane * 2;
  unsafeAtomicAdd(ap,     norm * v.x);
  unsafeAtomicAdd(ap + 1, norm * v.y);
}

__global__ void bias_relu_k(const float* __restrict__ agg, const float* __restrict__ b,
                            float* __restrict__ out, int n64) {
  int i = blockIdx.x * blockDim.x + threadIdx.x;
  if (i < n64) out[i] = fmaxf(agg[i] + b[i & 63], 0.f);
}

// ---------------- SE attention (scrambled .view semantics) ----------------
// flat[i] = concat[r = i/384, j = i%384] = H[j>>6][r*64 + (j&63)];
// channel c mean is over flat range [c*64N, (c+1)*64N).
__global__ void se_reduce_k(const float* __restrict__ H, float* __restrict__ spre,
                            unsigned nd /* 64*N */) {
  const unsigned c    = blockIdx.y;
  const unsigned base = c * nd;
  float local = 0.f;
  for (unsigned t = blockIdx.x * blockDim.x + threadIdx.x; t < nd;
       t += blockDim.x * gridDim.x) {
    unsigned i = base + t;
    unsigned r = i / 384u;
    unsigned j = i - r * 384u;
    local += H[(size_t)(j >> 6) * nd + r * 64u + (j & 63u)];
  }
  __shared__ float red[TPB];
  red[threadIdx.x] = local;
  __syncthreads();
  for (int s = TPB / 2; s > 0; s >>= 1) {
    if ((int)threadIdx.x < s) red[threadIdx.x] += red[threadIdx.x + s];
    __syncthreads();
  }
  if (threadIdx.x == 0) unsafeAtomicAdd(&spre[c], red[0]);
}

__global__ void se_mlp_k(const float* __restrict__ spre,
                         const float* __restrict__ fc1w, const float* __restrict__ fc1b,
                         const float* __restrict__ fc2w, const float* __restrict__ fc2b,
                         float* __restrict__ sact, float invCnt) {
  if (threadIdx.x == 0 && blockIdx.x == 0) {
    float m[6];
    for (int c = 0; c < 6; ++c) m[c] = spre[c] * invCnt;
    float h[30];
    for (int r = 0; r < 30; ++r) {
      float t = fc1b[r];
      for (int c = 0; c < 6; ++c) t += m[c] * fc1w[r * 6 + c];
      h[r] = fmaxf(t, 0.f);
    }
    for (int c = 0; c < 6; ++c) {
      float t = fc2b[c];
      for (int r = 0; r < 30; ++r) t += h[r] * fc2w[c * 30 + r];
      sact[c] = 1.f / (1.f + expf(-t));
    }
  }
}

// out[n,f] = conv_b + sum_c conv_w[c] * relu(s[c] * XM[0,c,f,n]),
// with XM[0,c,f,n] = flat[(c*64+f)*N + n] scrambled as above.
__global__ void final_k(const float* __restrict__ H, const float* __restrict__ sact,
                        const float* __restrict__ convw, const float* __restrict__ convb,
                        float* __restrict__ out, int N, unsigned nd) {
  int idx = blockIdx.x * blockDim.x + threadIdx.x;
  if (idx >= (int)nd) return;
  const unsigned f = (unsigned)idx & 63u;
  const unsigned n = (unsigned)idx >> 6;
  float acc = convb[0];
#pragma unroll
  for (unsigned c = 0; c < 6; ++c) {
    unsigned i = (c * 64u + f) * (unsigned)N + n;
    unsigned r = i / 384u;
    unsigned j = i - r * 384u;
    float v = H[(size_t)(j >> 6) * nd + r * 64u + (j & 63u)];
    acc += convw[c] * fmaxf(sact[c] * v, 0.f);
  }
  out[idx] = acc;
}

// ---------------- driver ----------------

extern "C" void kernel_launch(void* const* d_in, const int* in_sizes, int n_in,
                              void* d_out, int out_size, void* d_ws, size_t ws_size,
                              hipStream_t stream) {
  (void)n_in; (void)out_size; (void)ws_size;
  const float* x     = (const float*)d_in[0];
  const int*   ei[3] = {(const int*)d_in[1], (const int*)d_in[3], (const int*)d_in[5]};
  const float* ew[3] = {(const float*)d_in[2], (const float*)d_in[4], (const float*)d_in[6]};
  const float* W1    = (const float*)d_in[7];
  const float* b1    = (const float*)d_in[8];
  const float* W2    = (const float*)d_in[9];
  const float* b2    = (const float*)d_in[10];
  const float* fc1w  = (const float*)d_in[11];
  const float* fc1b  = (const float*)d_in[12];
  const float* fc2w  = (const float*)d_in[13];
  const float* fc2b  = (const float*)d_in[14];
  const float* convw = (const float*)d_in[15];
  const float* convb = (const float*)d_in[16];

  const int    N  = in_sizes[0] / 64;
  const int    E  = in_sizes[2];
  const size_t ND = (size_t)N * 64;

  // workspace layout (floats): H[6*ND] | dinv[3N] | hx[ND] | tmp[ND] | agg[ND] | spre[6] | sact[6]
  float* ws   = (float*)d_ws;
  float* Hb   = ws;
  float* dinv = Hb + 6 * ND;
  float* hx   = dinv + 3 * (size_t)N;
  float* tmp  = hx + ND;
  float* agg  = tmp + ND;
  float* spre = agg + ND;
  float* sact = spre + 6;

  const int eb    = (E + TPB - 1) / TPB;
  const int nb    = (int)((ND + TPB - 1) / TPB);
  const int n3b   = (3 * N + TPB - 1) / TPB;
  const int gemmb = (N + 31) / 32;
  const int edgeb = (E + 7) / 8;     // 8 wave32 warps per block, 1 edge per warp

  // symmetric-norm degrees (self-loop weight 1.0 folded into init)
  fill1_k<<<n3b, TPB, 0, stream>>>(dinv, 3 * N);
  for (int v = 0; v < 3; ++v)
    deg_acc_k<<<eb, TPB, 0, stream>>>(ei[v] + E, ew[v], dinv + (size_t)v * N, E);
  rsqrt_k<<<n3b, TPB, 0, stream>>>(dinv, 3 * N);

  // shared layer-1 GEMM: hx = x @ W1^T
  gemm_wmma_k<<<gemmb, TPB, 0, stream>>>(x, W1, hx, N);

  for (int v = 0; v < 3; ++v) {
    const float* dv = dinv + (size_t)v * N;
    float* h1 = Hb + (size_t)(2 * v) * ND;
    float* h2 = Hb + (size_t)(2 * v + 1) * ND;

    self_init_k<<<nb, TPB, 0, stream>>>(agg, hx, dv, (int)ND);
    edge_agg_k<<<edgeb, TPB, 0, stream>>>(ei[v], ei[v] + E, ew[v], dv, hx, agg, E);
    bias_relu_k<<<nb, TPB, 0, stream>>>(agg, b1, h1, (int)ND);

    gemm_wmma_k<<<gemmb, TPB, 0, stream>>>(h1, W2, tmp, N);

    self_init_k<<<nb, TPB, 0, stream>>>(agg, tmp, dv, (int)ND);
    edge_agg_k<<<edgeb, TPB, 0, stream>>>(ei[v], ei[v] + E, ew[v], dv, tmp, agg, E);
    bias_relu_k<<<nb, TPB, 0, stream>>>(agg, b2, h2, (int)ND);
  }

  // SE channel attention over the raw-reshaped concat
  zero_k<<<1, 32, 0, stream>>>(spre, 6);
  dim3 segrid(1024, 6, 1);
  se_reduce_k<<<segrid, TPB, 0, stream>>>(Hb, spre, (unsigned)ND);
  se_mlp_k<<<1, 32, 0, stream>>>(spre, fc1w, fc1b, fc2w, fc2b, sact,
                                 1.0f / (float)ND);

  final_k<<<nb, TPB, 0, stream>>>(Hb, sact, convw, convb, (float*)d_out, N,
                                  (unsigned)ND);
}